// MatrixReconstructionLayer_11897059410241
// MI455X (gfx1250) — compile-verified
//
#include <hip/hip_runtime.h>

#define FM_EPS 1e-8f

typedef float v4f __attribute__((ext_vector_type(4)));

// One block = 256 rows. Per row: load 32B (2 x b128, NT), compute 3x3 F
// analytically, stage 9 floats in LDS (stride-9 dwords: conflict-free,
// gcd(9,64)=1), then drain the 9216B tile to global with coalesced
// global_store_async_from_lds_b128 (ASYNCcnt path).
__global__ __launch_bounds__(256) void fundamental_matrix_kernel(
    const float* __restrict__ X, float* __restrict__ out, int n) {
  __shared__ float sOut[256 * 9];  // 9216 B staging tile

  const int tid  = threadIdx.x;
  const int base = (int)(blockIdx.x << 8);
  const int r    = base + tid;

  float F[9];
  if (r < n) {
    const v4f* __restrict__ X4 = (const v4f*)X;
    v4f a = __builtin_nontemporal_load(X4 + 2 * (size_t)r);      // f1 f2 rx ry
    v4f b = __builtin_nontemporal_load(X4 + 2 * (size_t)r + 1);  // rz tx ty tz
    const float f1 = a.x, f2 = a.y, rx = a.z, ry = a.w;
    const float rz = b.x, tx = b.y, ty = b.z, tz = b.w;

    float sx, cx, sy, cy, sz, cz;
    __sincosf(rx, &sx, &cx);
    __sincosf(ry, &sy, &cy);
    __sincosf(rz, &sz, &cz);

    // R = Rx @ (Ry @ Rz), expanded:
    const float r00 = cy * cz, r01 = -cy * sz, r02 = -sy;
    const float sxsy = sx * sy, cxsy = cx * sy;
    const float r10 = cx * sz - sxsy * cz;
    const float r11 = cx * cz + sxsy * sz;
    const float r12 = -sx * cy;
    const float r20 = sx * sz + cxsy * cz;
    const float r21 = sx * cz - cxsy * sz;
    const float r22 = cx * cy;

    // M = T @ R ; column j of M = t x R[:,j]
    const float m00 = ty * r20 - tz * r10;
    const float m01 = ty * r21 - tz * r11;
    const float m02 = ty * r22 - tz * r12;
    const float m10 = tz * r00 - tx * r20;
    const float m11 = tz * r01 - tx * r21;
    const float m12 = tz * r02 - tx * r22;
    const float m20 = tx * r10 - ty * r00;
    const float m21 = tx * r11 - ty * r01;
    const float m22 = tx * r12 - ty * r02;

    // F = diag(k2,k2,1) * M * diag(k1,k1,1)
    const float k1  = -__frcp_rn(f1 + FM_EPS);
    const float k2  = -__frcp_rn(f2 + FM_EPS);
    const float k12 = k1 * k2;

    F[0] = k12 * m00; F[1] = k12 * m01; F[2] = k2 * m02;
    F[3] = k12 * m10; F[4] = k12 * m11; F[5] = k2 * m12;
    F[6] = k1  * m20; F[7] = k1  * m21; F[8] = m22;

    #pragma unroll
    for (int j = 0; j < 9; ++j) sOut[tid * 9 + j] = F[j];
  }
  __syncthreads();

  if (base + 256 <= n) {
    // Full tile: 2304 floats = 576 x b128, coalesced LDS -> global async drain.
    // LDS address operands are DERIVED FROM &sOut (ptrtoint escape) so the
    // staging ds_store's cannot be dead-store-eliminated; low 32 bits of the
    // flat shared address are the wave-relative LDS byte address.
    const v4f* dst4 = (const v4f*)out + (size_t)blockIdx.x * 576;
    {
      const unsigned c0 = (unsigned)tid;
      const unsigned c1 = (unsigned)tid + 256u;
      const v4f* g0 = dst4 + c0;
      const v4f* g1 = dst4 + c1;
      unsigned l0 = (unsigned)(size_t)(&sOut[(size_t)c0 * 4]);
      unsigned l1 = (unsigned)(size_t)(&sOut[(size_t)c1 * 4]);
      asm volatile("global_store_async_from_lds_b128 %0, %1, off"
                   :: "v"(g0), "v"(l0) : "memory");
      asm volatile("global_store_async_from_lds_b128 %0, %1, off"
                   :: "v"(g1), "v"(l1) : "memory");
      if (tid < 64) {
        const unsigned c2 = (unsigned)tid + 512u;
        const v4f* g2 = dst4 + c2;
        unsigned l2 = (unsigned)(size_t)(&sOut[(size_t)c2 * 4]);
        asm volatile("global_store_async_from_lds_b128 %0, %1, off"
                     :: "v"(g2), "v"(l2) : "memory");
      }
    }
    asm volatile("s_wait_asynccnt 0x0" ::: "memory");
  } else if (r < n) {
    // Tail tile: plain per-row stores.
    float* o = out + (size_t)r * 9;
    #pragma unroll
    for (int j = 0; j < 9; ++j) o[j] = F[j];
  }
}

extern "C" void kernel_launch(void* const* d_in, const int* in_sizes, int n_in,
                              void* d_out, int out_size, void* d_ws, size_t ws_size,
                              hipStream_t stream) {
  (void)n_in; (void)out_size; (void)d_ws; (void)ws_size;
  const float* X = (const float*)d_in[0];
  float* out     = (float*)d_out;
  const int n    = in_sizes[0] / 8;            // rows of [B, 8]
  const int blocks = (n + 255) / 256;
  fundamental_matrix_kernel<<<blocks, 256, 0, stream>>>(X, out, n);
}